// MSC_59047210385981
// MI455X (gfx1250) — compile-verified
//
#include <hip/hip_runtime.h>

typedef float v2f __attribute__((ext_vector_type(2)));
typedef float v8f __attribute__((ext_vector_type(8)));

#define EPS_BN 1e-5f

// Problem sizes (fixed by setup_inputs)
#define Bn 8
#define Cc 16
#define Mc 8
#define HH 512
#define WW 512
#define NPIX (HH * WW)          // 262144
#define NV4  (NPIX / 4)         // 65536 float4 per channel

// MSP tile
#define TS 16
#define HALO 12
#define EXT (TS + 2 * HALO)     // 40
#define EXT2 (EXT * EXT)        // 1600

// Output layout (element offsets into float d_out)
#define OFF_FC   0ull
#define OFF_OFR  31457280ull
#define OFF_VALS 31457320ull
#define OFF_IDX  31457448ull
#define OFF_ORIG 31457576ull
#define OFF_CG   65012008ull

struct CP { const float *w, *g, *b, *m, *v; };
struct MSPParams {
  const float* x;
  CP c0, dc1, dc2, dc3, c40, c50, c4, c5, fd;
  const float *W1, *b1, *W2, *b2;
  float* cm;   // workspace: Bn * 256 floats, atomically accumulated
};

// ---------------- device helpers ----------------

__device__ __forceinline__ void lds_wave_fence() {
  // DS ops from one wave complete in order; dscnt==0 means all prior LDS
  // writes by this wave are visible to all its lanes.
  asm volatile("s_wait_dscnt 0" ::: "memory");
}

__device__ __forceinline__ float clip01(float x) {
  return fminf(fmaxf(x * (1.0f / 255.0f), 0.0f), 1.0f);
}

// CDNA5 hardware transcendentals: v_tanh_f32 / v_sin_f32 / v_cos_f32.
__device__ __forceinline__ float htanh(float x) {
#if __has_builtin(__builtin_amdgcn_tanhf)
  return __builtin_amdgcn_tanhf(x);
#elif __has_builtin(__builtin_amdgcn_tanh_f32)
  return __builtin_amdgcn_tanh_f32(x);
#else
  return tanhf(x);
#endif
}
__device__ __forceinline__ float hsin(float x) { return __sinf(x); }
__device__ __forceinline__ float hcos(float x) { return __cosf(x); }

// 1x1 conv (8->8) + BN + ReLU, weights uniform across lanes (scalar loads)
__device__ __forceinline__ void cbr8(const float* in, const CP& cp, float* out) {
#pragma unroll
  for (int mo = 0; mo < Mc; ++mo) {
    float acc = 0.f;
#pragma unroll
    for (int mi = 0; mi < Mc; ++mi) acc += cp.w[mo * Mc + mi] * in[mi];
    float sc = cp.g[mo] * rsqrtf(cp.v[mo] + EPS_BN);
    out[mo] = fmaxf(sc * (acc - cp.m[mo]) + cp.b[mo], 0.f);
  }
}

// D(16x16) = X(16x16) @ W(16x16) + bias, via four V_WMMA_F32_16X16X4_F32.
// X row-major [row=M][col=K], W row-major [K][N]. Lane mapping (wave32):
//   m = n = lane&15, half = lane>>4
//   A frag: a[j] = X[m][kk + half*2 + j]       (16x4 f32 A layout)
//   B frag: b[j] = W[kk + half*2 + j][n]       (4x16 f32 B layout)
//   C/D:    d[v] = D[v + 8*half][n]
__device__ __forceinline__ v8f wmma_gemm16(const float* __restrict__ X,
                                           const float* __restrict__ Wm,
                                           const float* __restrict__ bias,
                                           int m, int half) {
  const int n = m;
  v8f c;
  float bv = bias[n];
#pragma unroll
  for (int i = 0; i < 8; ++i) c[i] = bv;
#pragma unroll
  for (int kk = 0; kk < 16; kk += 4) {
    v2f a, b;
    a[0] = X[m * 16 + kk + half * 2 + 0];
    a[1] = X[m * 16 + kk + half * 2 + 1];
    b[0] = Wm[(kk + half * 2 + 0) * 16 + n];
    b[1] = Wm[(kk + half * 2 + 1) * 16 + n];
    c = __builtin_amdgcn_wmma_f32_16x16x4_f32(false, a, false, b, (short)0, c,
                                              false, false);
  }
  return c;
}

// One dilated branch: 3x3 dilated conv (8->8) + BN/ReLU, sin/cos mixing,
// four 1x1 cbr convs, then accumulate into the fd-conv partial sums.
template <int DIL>
__device__ __forceinline__ void do_branch(const MSPParams& P, const CP& dc,
                                          int bi, const float* __restrict__ x0s,
                                          int ey, int ex,
                                          float* __restrict__ xa) {
  float acc[Mc];
#pragma unroll
  for (int mo = 0; mo < Mc; ++mo) acc[mo] = 0.f;
#pragma unroll
  for (int ky = 0; ky < 3; ++ky)
#pragma unroll
    for (int kx = 0; kx < 3; ++kx) {
      const int yy = ey + (ky - 1) * DIL;
      const int xx = ex + (kx - 1) * DIL;
      const float* base = &x0s[yy * EXT + xx];
      float xv[Mc];
#pragma unroll
      for (int mi = 0; mi < Mc; ++mi) xv[mi] = base[mi * EXT2];
#pragma unroll
      for (int mo = 0; mo < Mc; ++mo) {
        const float* wr = &dc.w[(mo * Mc) * 9 + ky * 3 + kx];
#pragma unroll
        for (int mi = 0; mi < Mc; ++mi) acc[mo] += wr[mi * 9] * xv[mi];
      }
    }
  float uu[Mc], vv[Mc];
#pragma unroll
  for (int mo = 0; mo < Mc; ++mo) {
    float sc = dc.g[mo] * rsqrtf(dc.v[mo] + EPS_BN);
    float t = fmaxf(sc * (acc[mo] - dc.m[mo]) + dc.b[mo], 0.f);
    uu[mo] = t * hcos(t);   // xi * cos(xi)
    vv[mo] = t * hsin(t);   // xi * sin(xi)
  }
  float cc[Mc], ss[Mc];
  cbr8(uu, P.c50, cc);      // c = cbr(xi*cos(xi), c5_0)
  cbr8(vv, P.c40, ss);      // s = cbr(xi*sin(xi), c4_0)
  float pm[Mc], qm[Mc];
#pragma unroll
  for (int mi = 0; mi < Mc; ++mi) {
    pm[mi] = ss[mi] * cc[mi];
    qm[mi] = ss[mi] + cc[mi];
  }
  float t1[Mc], t2[Mc];
  cbr8(pm, P.c4, t1);       // cbr(s*c, c_4)
  cbr8(qm, P.c5, t2);       // cbr(s+c, c_5)
#pragma unroll
  for (int oc = 0; oc < 16; ++oc) {
    float a2 = 0.f;
#pragma unroll
    for (int mi = 0; mi < Mc; ++mi)
      a2 += P.fd.w[oc * 32 + bi * 8 + mi] * (t1[mi] + t2[mi]);
    xa[oc] += a2;
  }
}

// ---------------- kernel 0: zero cm accumulator ----------------

__global__ void zero_cm_kernel(float* cm) {
  int i = blockIdx.x * 256 + threadIdx.x;
  if (i < Bn * 256) cm[i] = 0.f;
}

// ---------------- kernel 1: fused MSP + h-GEMMs + Gram ----------------

__global__ __launch_bounds__(256) void msp_cm_kernel(MSPParams P) {
  __shared__ float x0s[Mc * EXT2];     // 51200 B; later aliased as xa_s[256*16]
  __shared__ float hbuf[8 * 256];      // per-wave 16x16 h scratch
  __shared__ float w1s[256], w2s[256];
  __shared__ float b1s[16], b2s[16];

  const int tid = threadIdx.x;
  const int blk = blockIdx.x;
  const int b = blk >> 10;             // 1024 tiles (32x32) per batch
  const int tile = blk & 1023;
  const int oy = (tile >> 5) * TS - HALO;
  const int ox = (tile & 31) * TS - HALO;

  w1s[tid] = P.W1[tid];
  w2s[tid] = P.W2[tid];
  if (tid < 16) { b1s[tid] = P.b1[tid]; b2s[tid] = P.b2[tid]; }

  // ---- Stage A: x0 = cbr1x1(clip(x/255)) on the haloed tile (zero-pad OOB)
  float s0[Mc], t0[Mc];
#pragma unroll
  for (int mo = 0; mo < Mc; ++mo) {
    float sc = P.c0.g[mo] * rsqrtf(P.c0.v[mo] + EPS_BN);
    s0[mo] = sc;
    t0[mo] = P.c0.b[mo] - sc * P.c0.m[mo];
  }
  const float* xb = P.x + (size_t)b * Cc * NPIX;
  for (int i = tid; i < EXT2; i += 256) {
    int ey = i / EXT, ex = i - ey * EXT;
    int gy = oy + ey, gx = ox + ex;
    float outv[Mc];
    if ((unsigned)gy < (unsigned)HH && (unsigned)gx < (unsigned)WW) {
      float acc[Mc];
#pragma unroll
      for (int mo = 0; mo < Mc; ++mo) acc[mo] = 0.f;
      const float* px = xb + (size_t)gy * WW + gx;
#pragma unroll
      for (int c = 0; c < Cc; ++c) {
        float xv = clip01(px[(size_t)c * NPIX]);
#pragma unroll
        for (int mo = 0; mo < Mc; ++mo) acc[mo] += P.c0.w[mo * Cc + c] * xv;
      }
#pragma unroll
      for (int mo = 0; mo < Mc; ++mo)
        outv[mo] = fmaxf(s0[mo] * acc[mo] + t0[mo], 0.f);
    } else {
#pragma unroll
      for (int mo = 0; mo < Mc; ++mo) outv[mo] = 0.f;
    }
#pragma unroll
    for (int mo = 0; mo < Mc; ++mo) x0s[mo * EXT2 + i] = outv[mo];
  }
  __syncthreads();

  // ---- Stage B: one pixel per thread; 3 dilated branches; accumulate fd conv
  const int py = tid >> 4, px0 = tid & 15;
  const int ey = py + HALO, ex = px0 + HALO;

  float xa[16];
#pragma unroll
  for (int oc = 0; oc < 16; ++oc) xa[oc] = 0.f;

  do_branch<1>(P, P.dc1, 0, x0s, ey, ex, xa);
  do_branch<6>(P, P.dc2, 1, x0s, ey, ex, xa);
  do_branch<12>(P, P.dc3, 2, x0s, ey, ex, xa);

  // x0's slot in the concat (channels 24..31)
  {
    float x0v[Mc];
#pragma unroll
    for (int mi = 0; mi < Mc; ++mi) x0v[mi] = x0s[mi * EXT2 + ey * EXT + ex];
#pragma unroll
    for (int oc = 0; oc < 16; ++oc) {
      float a2 = 0.f;
#pragma unroll
      for (int mi = 0; mi < Mc; ++mi) a2 += P.fd.w[oc * 32 + 24 + mi] * x0v[mi];
      xa[oc] += a2;
    }
  }
  // fd BN + ReLU
#pragma unroll
  for (int oc = 0; oc < 16; ++oc) {
    float sc = P.fd.g[oc] * rsqrtf(P.fd.v[oc] + EPS_BN);
    xa[oc] = fmaxf(sc * (xa[oc] - P.fd.m[oc]) + P.fd.b[oc], 0.f);
  }

  __syncthreads();                 // everyone done reading x0s
  float* xa_s = x0s;               // alias: 256 pixels x 16 channels (16 KB)
#pragma unroll
  for (int oc = 0; oc < 16; ++oc) xa_s[tid * 16 + oc] = xa[oc];
  __syncthreads();

  // ---- Stage C: per-wave WMMA; 16 pixel-groups, 8 waves -> 2 groups each
  const int wid = tid >> 5, lane = tid & 31;
  const int m = lane & 15, half = lane >> 4;
  float* hb = &hbuf[wid * 256];

  v8f cmacc;
#pragma unroll
  for (int i = 0; i < 8; ++i) cmacc[i] = 0.f;

  for (int g = wid * 2; g < wid * 2 + 2; ++g) {
    const float* Xg = &xa_s[g * 256];

    v8f h1 = wmma_gemm16(Xg, w1s, b1s, m, half);   // xa @ W1 + b1
#pragma unroll
    for (int i = 0; i < 8; ++i) h1[i] = htanh(h1[i]);
#pragma unroll
    for (int v = 0; v < 8; ++v) hb[(v + 8 * half) * 16 + m] = h1[v];
    lds_wave_fence();

    v8f h2 = wmma_gemm16(hb, w2s, b2s, m, half);   // h1 @ W2 + b2
#pragma unroll
    for (int i = 0; i < 8; ++i) h2[i] = htanh(h2[i]);
#pragma unroll
    for (int v = 0; v < 8; ++v) hb[(v + 8 * half) * 16 + m] = h2[v];
    lds_wave_fence();

    // Gram: cm += H^T H  (A[c][k] = H[k][c], B[k][d] = H[k][d]; k = pixel)
#pragma unroll
    for (int pp = 0; pp < 16; pp += 4) {
      v2f a, bb;
      a[0] = hb[(pp + half * 2 + 0) * 16 + m];
      a[1] = hb[(pp + half * 2 + 1) * 16 + m];
      bb[0] = a[0];
      bb[1] = a[1];
      cmacc = __builtin_amdgcn_wmma_f32_16x16x4_f32(false, a, false, bb,
                                                    (short)0, cmacc, false, false);
    }
    lds_wave_fence();
  }

  float* cmg = P.cm + b * 256;
#pragma unroll
  for (int v = 0; v < 8; ++v)
    atomicAdd(&cmg[(v + 8 * half) * 16 + m], cmacc[v]);
}

// ---------------- kernel 2: per-batch finalize ----------------

__global__ void finalize_kernel(const float* __restrict__ cm_g,
                                float* __restrict__ out,
                                float* __restrict__ wv_ws,
                                int* __restrict__ idx_ws) {
  const int b = blockIdx.x;
  if (threadIdx.x != 0) return;

  float cmv[16][16];
  for (int c = 0; c < 16; ++c)
    for (int d = 0; d < 16; ++d) cmv[c][d] = cm_g[b * 256 + c * 16 + d];
  for (int c = 0; c < 16; ++c) cmv[c][c] = 0.f;
  for (int c = 0; c < 16; ++c) {
    float ss = 0.f;
    for (int d = 0; d < 16; ++d) ss += cmv[c][d] * cmv[c][d];
    float inv = 1.f / fmaxf(sqrtf(ss), 1e-12f);
    for (int d = 0; d < 16; ++d) cmv[c][d] *= inv;
    cmv[c][c] = 0.f;
  }
  float wv[16];
  for (int c = 0; c < 16; ++c) {
    float s = 0.f;
    for (int d = 0; d < 16; ++d) s += fabsf(cmv[c][d]);
    wv[c] = s;
  }
  // stable descending argsort (matches jnp.argsort(-w) tie order)
  int ord[16];
  for (int i = 0; i < 16; ++i) ord[i] = i;
  for (int i = 1; i < 16; ++i) {
    int key = ord[i];
    float kv = wv[key];
    int j = i - 1;
    while (j >= 0 && wv[ord[j]] < kv) { ord[j + 1] = ord[j]; --j; }
    ord[j + 1] = key;
  }
  float* vals_o = out + OFF_VALS;
  float* idx_o = out + OFF_IDX;
  float* ofr_o = out + OFF_OFR;
  for (int i = 0; i < 16; ++i) {
    vals_o[b * 16 + i] = wv[ord[i]];
    idx_o[b * 16 + i] = (float)ord[i];
  }
  for (int k = 0; k < 5; ++k)
    ofr_o[k * 8 + b] = (wv[ord[3 * k]] + wv[ord[3 * k + 1]] + wv[ord[3 * k + 2]]) *
                       (1.f / 3.f);
  for (int c = 0; c < 16; ++c) wv_ws[b * 16 + c] = wv[c];
  for (int i = 0; i < 16; ++i) idx_ws[b * 16 + i] = ord[i];
}

// ---------------- kernel 3: x_orig + x_cg (streaming, float4) ----------------

__global__ void xcg_kernel(const float* __restrict__ x,
                           const float* __restrict__ wv,
                           float* __restrict__ out) {
  const unsigned long long TOT = (unsigned long long)Bn * Cc * NV4;
  const float4* xv = (const float4*)x;
  float4* o_orig = (float4*)(out + OFF_ORIG);
  float4* o_cg = (float4*)(out + OFF_CG);
  for (unsigned long long i =
           (unsigned long long)blockIdx.x * blockDim.x + threadIdx.x;
       i < TOT; i += (unsigned long long)gridDim.x * blockDim.x) {
    unsigned long long cl = i >> 16;       // b*16 + c
    float w = wv[cl];
    float4 v = xv[i];
    float4 o;
    o.x = clip01(v.x); o.y = clip01(v.y); o.z = clip01(v.z); o.w = clip01(v.w);
    o_orig[i] = o;
    float4 g;
    g.x = o.x * w; g.y = o.y * w; g.z = o.z * w; g.w = o.w * w;
    o_cg[i] = g;
  }
}

// ---------------- kernel 4: fc gather of raw x channels ----------------

__global__ void fc_kernel(const float* __restrict__ x,
                          const int* __restrict__ idx_ws,
                          float* __restrict__ out) {
  const unsigned long long TOT = (unsigned long long)Bn * 15 * NV4;
  const float4* xv = (const float4*)x;
  float4* o_fc = (float4*)(out + OFF_FC);
  for (unsigned long long i =
           (unsigned long long)blockIdx.x * blockDim.x + threadIdx.x;
       i < TOT; i += (unsigned long long)gridDim.x * blockDim.x) {
    unsigned long long tl = i >> 16;       // b*15 + t
    unsigned long long j = i & 65535ull;
    int b = (int)(tl / 15);
    int t = (int)(tl - (unsigned long long)b * 15);
    int ch = idx_ws[b * 16 + t];
    o_fc[(tl << 16) + j] = xv[(((unsigned long long)(b * 16 + ch)) << 16) + j];
  }
}

// ---------------- launch ----------------

extern "C" void kernel_launch(void* const* d_in, const int* in_sizes, int n_in,
                              void* d_out, int out_size, void* d_ws,
                              size_t ws_size, hipStream_t stream) {
  const float* x = (const float*)d_in[0];
  auto cp = [&](int base) {
    CP c;
    c.w = (const float*)d_in[base + 0];
    c.g = (const float*)d_in[base + 1];
    c.b = (const float*)d_in[base + 2];
    c.m = (const float*)d_in[base + 3];
    c.v = (const float*)d_in[base + 4];
    return c;
  };
  MSPParams P;
  P.x = x;
  P.c0 = cp(1);
  P.dc1 = cp(6);
  P.dc2 = cp(11);
  P.dc3 = cp(16);
  P.c40 = cp(21);   // 'c4_0'
  P.c50 = cp(26);   // 'c5_0'
  P.c4 = cp(31);    // 'c_4'
  P.c5 = cp(36);    // 'c_5'
  P.fd = cp(41);
  P.W1 = (const float*)d_in[46];
  P.b1 = (const float*)d_in[47];
  P.W2 = (const float*)d_in[48];
  P.b2 = (const float*)d_in[49];

  float* cm = (float*)d_ws;              // Bn*256 floats
  float* wv = cm + Bn * 256;             // Bn*16 floats
  int* idxw = (int*)(wv + Bn * 16);      // Bn*16 ints
  P.cm = cm;

  float* out = (float*)d_out;

  zero_cm_kernel<<<8, 256, 0, stream>>>(cm);
  msp_cm_kernel<<<dim3(Bn * 1024), 256, 0, stream>>>(P);
  finalize_kernel<<<Bn, 32, 0, stream>>>(cm, out, wv, idxw);
  xcg_kernel<<<8192, 256, 0, stream>>>(x, wv, out);
  fc_kernel<<<8192, 256, 0, stream>>>(x, idxw, out);
}